// SegmentDecoderV2_72834055406375
// MI455X (gfx1250) — compile-verified
//
#include <hip/hip_runtime.h>
#include <hip/hip_bf16.h>

#define NN 12288          // num nodes
#define DD 128            // embedding dim
#define NT (NN / 16)      // 768 tiles per output dimension
#define WAVES 8           // waves (tiles) per workgroup

typedef __attribute__((ext_vector_type(2))) float v2f;
typedef __attribute__((ext_vector_type(4))) float v4f;
typedef __attribute__((ext_vector_type(8))) float v8f;

__global__ __launch_bounds__(256)
void SegmentDecoderV2_tiles(const float* __restrict__ z1,
                            const float* __restrict__ z2,
                            const int*   __restrict__ cls,
                            const int*   __restrict__ batch,
                            float*       __restrict__ out) {
  // Force wave id scalar so tile coords / early-exit branch are SALU and
  // EXEC stays all-ones for the WMMA path.
  const int wave = __builtin_amdgcn_readfirstlane((int)(threadIdx.x >> 5));
  const int lane = (int)(threadIdx.x & 31);
  const int tile = (int)blockIdx.x * WAVES + wave;
  const int ty = tile / NT;          // row-tile index
  const int tx = tile - ty * NT;     // col-tile index
  const int r0 = ty << 4;
  const int c0 = tx << 4;

  // batch is sorted: the tile can contain nonzeros only if the graph-id
  // ranges of the 16 rows and 16 cols overlap.
  const int rlo = batch[r0];
  const int rhi = batch[r0 + 15];
  const int clo = batch[c0];
  const int chi = batch[c0 + 15];

  if (rlo > chi || clo > rhi) {
    // All-zero tile: streaming zero-fill. Each lane covers half a row with
    // two contiguous b128 stores (32 lanes * 8 floats = 256 floats).
    const int zr = r0 + (lane >> 1);
    const int zc = c0 + ((lane & 1) << 3);
    v4f zero4 = {0.0f, 0.0f, 0.0f, 0.0f};
    v4f* p = (v4f*)(out + (size_t)zr * NN + zc);
    p[0] = zero4;
    p[1] = zero4;
    return;
  }

  // ---- FP32 WMMA 16x16x4, K = 128 in steps of 4 ----
  // A (16x4) lane layout: lanes 0-15 hold M=lane, K = k..k+1 in v[0..1];
  // lanes 16-31 hold the same M rows, K = k+2..k+3.
  // B (4x16) mirrors with N = lane&15. For z1 @ z2^T, B[k][n] = z2[c0+n][k],
  // so both operands are contiguous-K b64 loads from each matrix's rows.
  const int m  = lane & 15;
  const int kh = (lane >> 4) << 1;   // 0 for low half-wave, 2 for high
  const float* ap = z1 + (size_t)(r0 + m) * DD + kh;
  const float* bp = z2 + (size_t)(c0 + m) * DD + kh;

  v8f acc = {};
#pragma unroll
  for (int k = 0; k < DD; k += 4) {
    v2f a = *(const v2f*)(ap + k);
    v2f b = *(const v2f*)(bp + k);
    // 8 args: (neg_a, A, neg_b, B, c_mod, C, reuse_a, reuse_b)
    acc = __builtin_amdgcn_wmma_f32_16x16x4_f32(
        false, a, false, b, (short)0, acc, false, false);
  }

  // ---- per-element mask + store ----
  // C/D layout: VGPR v holds M = v (lanes 0-15) and M = v+8 (lanes 16-31),
  // N = lane & 15.
  const int col  = c0 + m;
  const int bcol = batch[col];
  const int ccol = cls[col];
  const int roff = r0 + ((lane >> 4) << 3);

#pragma unroll
  for (int v = 0; v < 8; ++v) {
    const int row = roff + v;
    const int br  = batch[row];
    const int cr  = cls[row];
    // classes are in [0, 27): the {24,25,26} exclusion reduces to cr >= 24
    const bool keep = (br == bcol) && (cr == ccol) && (cr < 24) && (row != col);
    out[(size_t)row * NN + col] = keep ? acc[v] : 0.0f;
  }
}

extern "C" void kernel_launch(void* const* d_in, const int* in_sizes, int n_in,
                              void* d_out, int out_size, void* d_ws, size_t ws_size,
                              hipStream_t stream) {
  const float* z1    = (const float*)d_in[0];
  const float* z2    = (const float*)d_in[1];
  const int*   cls   = (const int*)d_in[2];
  const int*   batch = (const int*)d_in[3];
  float*       out   = (float*)d_out;

  const int total_tiles = NT * NT;              // 589824
  const int blocks = total_tiles / WAVES;       // 73728
  SegmentDecoderV2_tiles<<<blocks, 32 * WAVES, 0, stream>>>(z1, z2, cls, batch, out);
}